// CLIPAttention_27401891348806
// MI455X (gfx1250) — compile-verified
//
#include <hip/hip_runtime.h>
#include <cstddef>

// ---------------- problem constants (from setup_inputs) ----------------
constexpr int Bb = 4;
constexpr int Hh = 16;
constexpr int DH = 64;
constexpr int Cc = 1024;     // Hh * DH
constexpr int Mm = 1;
constexpr int Nn = 16;
constexpr int Ll = 256;
constexpr int Tt = Mm + Nn * Ll;   // 4097
constexpr int ROWS = Bb * Tt;      // 16388
constexpr int SSTRIDE = 261;       // 257 cols padded -> conflict-free banks
constexpr float QSCALE = 0.125f;   // dh^-0.5 = 64^-0.5

typedef __attribute__((ext_vector_type(2))) float v2f;
typedef __attribute__((ext_vector_type(8))) float v8f;

__device__ __forceinline__ v2f ld2g(const float* p) {
    float2 t = *(const float2*)p;
    v2f r; r[0] = t.x; r[1] = t.y;
    return r;
}

__device__ __forceinline__ v8f wmma_f32(v2f a, v2f b, v8f c) {
    // D(16x16,f32) = A(16x4,f32) x B(4x16,f32) + C
    return __builtin_amdgcn_wmma_f32_16x16x4_f32(false, a, false, b, (short)0, c,
                                                 false, false);
}

// ---------------- register K-chunk for the 16x64 GEMM strip ------------
// 4 WMMA K-steps (16 K values): A tile + 4 B tiles, all independent loads.
struct KChunk {
    v2f a[4];
    v2f b0[4], b1[4], b2[4], b3[4];
};

__device__ __forceinline__ void load_chunk(KChunk& c, const float* aptr,
                                           const float* b0p, const float* b1p,
                                           const float* b2p, const float* b3p,
                                           int kbase, int koff)
{
    #pragma unroll
    for (int s = 0; s < 4; ++s) {
        const int kk = kbase + 4 * s + koff;
        c.a[s]  = ld2g(aptr + kk);
        c.b0[s] = ld2g(b0p + kk);
        c.b1[s] = ld2g(b1p + kk);
        c.b2[s] = ld2g(b2p + kk);
        c.b3[s] = ld2g(b3p + kk);
    }
}

__device__ __forceinline__ void mma_chunk(const KChunk& c, v8f& d0, v8f& d1,
                                          v8f& d2, v8f& d3)
{
    #pragma unroll
    for (int s = 0; s < 4; ++s) {
        d0 = wmma_f32(c.a[s], c.b0[s], d0);
        d1 = wmma_f32(c.a[s], c.b1[s], d1);
        d2 = wmma_f32(c.a[s], c.b2[s], d2);
        d3 = wmma_f32(c.a[s], c.b3[s], d3);
    }
}

// Software-pipelined 16-row x 64-col strip GEMM: D = X[rows] . W[cols]^T
__device__ __forceinline__ void strip_gemm(const float* aptr,
                                           const float* b0p, const float* b1p,
                                           const float* b2p, const float* b3p,
                                           int koff,
                                           v8f& d0, v8f& d1, v8f& d2, v8f& d3)
{
    KChunk c0, c1;
    load_chunk(c0, aptr, b0p, b1p, b2p, b3p, 0, koff);
    for (int kb = 0; kb < Cc; kb += 32) {
        load_chunk(c1, aptr, b0p, b1p, b2p, b3p, kb + 16, koff);   // prefetch
        mma_chunk(c0, d0, d1, d2, d3);
        if (kb + 32 < Cc)
            load_chunk(c0, aptr, b0p, b1p, b2p, b3p, kb + 32, koff); // prefetch
        mma_chunk(c1, d0, d1, d2, d3);
    }
}

// ---------------------------------------------------------------------
// Kernel 1: fused Q/K/V projection.
// One wave computes a 16-row x 64-col output strip of one of Q/K/V.
// Output layout: (B, H, T, dh).
// ---------------------------------------------------------------------
__global__ __launch_bounds__(32) void qkv_proj_kernel(
    const float* __restrict__ X,
    const float* __restrict__ Wq, const float* __restrict__ bq,
    const float* __restrict__ Wk, const float* __restrict__ bk,
    const float* __restrict__ Wv, const float* __restrict__ bv,
    float* __restrict__ qo, float* __restrict__ ko, float* __restrict__ vo)
{
    const int lane  = threadIdx.x;
    const int row0  = blockIdx.y * 16;
    const int col0  = blockIdx.x * 64;
    const int which = blockIdx.z;

    const float* W; const float* bias; float* out; float scale;
    if (which == 0)      { W = Wq; bias = bq; out = qo; scale = QSCALE; }
    else if (which == 1) { W = Wk; bias = bk; out = ko; scale = 1.0f;  }
    else                 { W = Wv; bias = bv; out = vo; scale = 1.0f;  }

    const int nlane = lane & 15;
    const int koff  = 2 * (lane >> 4);   // lanes 0-15: K 0,1 ; lanes 16-31: K 2,3
    const int mbase = 8 * (lane >> 4);

    int ar = row0 + nlane;
    if (ar > ROWS - 1) ar = ROWS - 1;    // clamp loads; stores are guarded
    const float* aptr = X + (size_t)ar * Cc;
    const float* b0p  = W + (size_t)(col0 +  0 + nlane) * Cc;
    const float* b1p  = W + (size_t)(col0 + 16 + nlane) * Cc;
    const float* b2p  = W + (size_t)(col0 + 32 + nlane) * Cc;
    const float* b3p  = W + (size_t)(col0 + 48 + nlane) * Cc;

    v8f d0 = {}, d1 = {}, d2 = {}, d3 = {};
    strip_gemm(aptr, b0p, b1p, b2p, b3p, koff, d0, d1, d2, d3);

    #pragma unroll
    for (int j = 0; j < 4; ++j) {
        v8f acc = (j == 0) ? d0 : (j == 1) ? d1 : (j == 2) ? d2 : d3;
        const int c = col0 + j * 16 + nlane;
        const float bc = bias[c];
        const int h = c >> 6;
        const int d = c & 63;
        #pragma unroll
        for (int r = 0; r < 8; ++r) {
            const int gr = row0 + r + mbase;
            if (gr < ROWS) {
                const int bidx = gr / Tt;
                const int tt   = gr % Tt;
                out[(((size_t)(bidx * Hh + h)) * Tt + tt) * DH + d] =
                    (acc[r] + bc) * scale;
            }
        }
    }
}

// ---------------------------------------------------------------------
// Kernel 2: local windowed attention. One 256-thread block per (b,h,n).
// S (256 x 257) lives in dynamic LDS with stride 261.
// ---------------------------------------------------------------------
__global__ __launch_bounds__(256) void local_attn_kernel(
    const float* __restrict__ q, const float* __restrict__ k,
    const float* __restrict__ v, float* __restrict__ o)
{
    extern __shared__ float S[];  // 256 * SSTRIDE floats

    const int w = blockIdx.x;
    const int b = w / (Hh * Nn);
    const int h = (w / Nn) % Hh;
    const int n = w % Nn;

    const size_t head_off = ((size_t)(b * Hh + h)) * Tt * DH;
    const int tq = Mm + n * Ll;
    const float* ql = q + head_off + (size_t)tq * DH;
    const float* kh = k + head_off;                    // kg = kh row 0
    const float* kl = kh + (size_t)tq * DH;
    const float* vh = v + head_off;                    // vg = vh row 0
    const float* vl = vh + (size_t)tq * DH;

    const int tid   = threadIdx.x;
    const int wv    = tid >> 5;
    const int lane  = tid & 31;
    const int nlane = lane & 15;
    const int koff  = 2 * (lane >> 4);
    const int mbase = 8 * (lane >> 4);

    // ---- Phase 1: S[l][c] = ql . kl^T (q pre-scaled), 16x16 tiles --------
    // All 32 loads hoisted ahead of the WMMA chain (independent registers).
    for (int tile = wv; tile < 256; tile += 8) {
        const int rt = tile >> 4, ct = tile & 15;
        const float* arow = ql + (size_t)(rt * 16 + nlane) * DH;
        const float* brow = kl + (size_t)(ct * 16 + nlane) * DH;
        v2f av[16], bv[16];
        #pragma unroll
        for (int it = 0; it < 16; ++it) {
            const int kk = it * 4 + koff;
            av[it] = ld2g(arow + kk);
            bv[it] = ld2g(brow + kk);
        }
        v8f acc = {};
        #pragma unroll
        for (int it = 0; it < 16; ++it) acc = wmma_f32(av[it], bv[it], acc);
        #pragma unroll
        for (int r = 0; r < 8; ++r) {
            const int row = rt * 16 + r + mbase;
            S[row * SSTRIDE + ct * 16 + nlane] = acc[r];
        }
    }
    // global-key column (col 256): sg[l] = ql[l] . kg
    {
        float acc = 0.f;
        const float* arow = ql + (size_t)tid * DH;
        #pragma unroll
        for (int i = 0; i < DH; ++i) acc += arow[i] * kh[i];
        S[tid * SSTRIDE + 256] = acc;
    }
    __syncthreads();

    // ---- Phase 2: row-wise softmax over 257 entries (1 thread / row) -----
    {
        float* row = S + (size_t)tid * SSTRIDE;
        float m = -3.0e38f;
        for (int c = 0; c < 257; ++c) m = fmaxf(m, row[c]);
        float sum = 0.f;
        for (int c = 0; c < 257; ++c) {
            const float p = __expf(row[c] - m);
            row[c] = p;
            sum += p;
        }
        const float inv = 1.0f / sum;
        for (int c = 0; c < 257; ++c) row[c] *= inv;
    }
    __syncthreads();

    // ---- Phase 3: O = pl @ vl + pg (x) vg, pipelined LDS/global loads ----
    for (int tile = wv; tile < 64; tile += 8) {
        const int rt = tile >> 2, dt = tile & 3;
        const int d  = dt * 16 + nlane;
        const float* arow = S + (size_t)(rt * 16 + nlane) * SSTRIDE;

        v2f pa0[8], pb0[8], pa1[8], pb1[8];
        auto load_pc = [&](v2f* pa, v2f* pb, int kbase) {
            #pragma unroll
            for (int s = 0; s < 8; ++s) {
                const int kk = kbase + 4 * s + koff;
                v2f a; a[0] = arow[kk]; a[1] = arow[kk + 1];
                v2f bb; bb[0] = vl[(size_t)kk * DH + d];
                        bb[1] = vl[(size_t)(kk + 1) * DH + d];
                pa[s] = a; pb[s] = bb;
            }
        };
        v8f acc = {};
        load_pc(pa0, pb0, 0);
        #pragma unroll
        for (int kb = 0; kb < Ll; kb += 64) {
            load_pc(pa1, pb1, kb + 32);                 // prefetch
            #pragma unroll
            for (int s = 0; s < 8; ++s) acc = wmma_f32(pa0[s], pb0[s], acc);
            if (kb + 64 < Ll) load_pc(pa0, pb0, kb + 64); // prefetch
            #pragma unroll
            for (int s = 0; s < 8; ++s) acc = wmma_f32(pa1[s], pb1[s], acc);
        }

        const float vg = vh[d];
        #pragma unroll
        for (int r = 0; r < 8; ++r) {
            const int row = rt * 16 + r + mbase;
            const float val = acc[r] + S[row * SSTRIDE + 256] * vg;
            const int tgl = tq + row;
            o[((size_t)b * Tt + tgl) * Cc + h * DH + d] = val;
        }
    }
}

// ---------------------------------------------------------------------
// Kernel 3: CLS-token (global) attention. One block per (b,h). Tiny.
// ---------------------------------------------------------------------
__global__ __launch_bounds__(256) void cls_attn_kernel(
    const float* __restrict__ q, const float* __restrict__ k,
    const float* __restrict__ v, float* __restrict__ o)
{
    __shared__ float s[Tt];
    __shared__ float qsh[DH];
    __shared__ float red[256];
    __shared__ float ored[256];

    const int bh = blockIdx.x;
    const int b = bh >> 4, h = bh & 15;
    const size_t head_off = ((size_t)(b * Hh + h)) * Tt * DH;
    const float* kh = k + head_off;
    const float* vh = v + head_off;
    const int tid = threadIdx.x;

    if (tid < DH) qsh[tid] = q[head_off + tid];  // qg (already scaled)
    __syncthreads();

    float lmax = -3.0e38f;
    for (int t = tid; t < Tt; t += 256) {
        float acc = 0.f;
        const float* kr = kh + (size_t)t * DH;
        #pragma unroll
        for (int i = 0; i < DH; ++i) acc += kr[i] * qsh[i];
        s[t] = acc;
        lmax = fmaxf(lmax, acc);
    }
    red[tid] = lmax; __syncthreads();
    for (int st = 128; st > 0; st >>= 1) {
        if (tid < st) red[tid] = fmaxf(red[tid], red[tid + st]);
        __syncthreads();
    }
    const float m = red[0];
    __syncthreads();

    float lsum = 0.f;
    for (int t = tid; t < Tt; t += 256) {
        const float p = __expf(s[t] - m);
        s[t] = p;
        lsum += p;
    }
    red[tid] = lsum; __syncthreads();
    for (int st = 128; st > 0; st >>= 1) {
        if (tid < st) red[tid] += red[tid + st];
        __syncthreads();
    }
    const float inv = 1.0f / red[0];
    __syncthreads();

    const int d = tid & 63, g = tid >> 6;
    float accd = 0.f;
    for (int t = g; t < Tt; t += 4) accd += s[t] * vh[(size_t)t * DH + d];
    ored[tid] = accd;
    __syncthreads();
    if (tid < DH) {
        const float val =
            (ored[tid] + ored[tid + 64] + ored[tid + 128] + ored[tid + 192]) * inv;
        o[((size_t)b * Tt + 0) * Cc + h * DH + tid] = val;
    }
}

// ---------------------------------------------------------------------
// Kernel 4: output projection  out = AO @ Wo^T + bo  (row-major B,T,C)
// ---------------------------------------------------------------------
__global__ __launch_bounds__(32) void out_proj_kernel(
    const float* __restrict__ X, const float* __restrict__ Wo,
    const float* __restrict__ bo, float* __restrict__ out)
{
    const int lane  = threadIdx.x;
    const int row0  = blockIdx.y * 16;
    const int col0  = blockIdx.x * 64;
    const int nlane = lane & 15;
    const int koff  = 2 * (lane >> 4);
    const int mbase = 8 * (lane >> 4);

    int ar = row0 + nlane;
    if (ar > ROWS - 1) ar = ROWS - 1;
    const float* aptr = X + (size_t)ar * Cc;
    const float* b0p  = Wo + (size_t)(col0 +  0 + nlane) * Cc;
    const float* b1p  = Wo + (size_t)(col0 + 16 + nlane) * Cc;
    const float* b2p  = Wo + (size_t)(col0 + 32 + nlane) * Cc;
    const float* b3p  = Wo + (size_t)(col0 + 48 + nlane) * Cc;

    v8f d0 = {}, d1 = {}, d2 = {}, d3 = {};
    strip_gemm(aptr, b0p, b1p, b2p, b3p, koff, d0, d1, d2, d3);

    #pragma unroll
    for (int j = 0; j < 4; ++j) {
        v8f acc = (j == 0) ? d0 : (j == 1) ? d1 : (j == 2) ? d2 : d3;
        const int c = col0 + j * 16 + nlane;
        const float bc = bo[c];
        #pragma unroll
        for (int r = 0; r < 8; ++r) {
            const int gr = row0 + r + mbase;
            if (gr < ROWS) out[(size_t)gr * Cc + c] = acc[r] + bc;
        }
    }
}

// ---------------------------------------------------------------------
extern "C" void kernel_launch(void* const* d_in, const int* in_sizes, int n_in,
                              void* d_out, int out_size, void* d_ws, size_t ws_size,
                              hipStream_t stream)
{
    (void)in_sizes; (void)n_in; (void)out_size; (void)ws_size;

    const float* X  = (const float*)d_in[0];
    const float* Wq = (const float*)d_in[1]; const float* bq = (const float*)d_in[2];
    const float* Wk = (const float*)d_in[3]; const float* bk = (const float*)d_in[4];
    const float* Wv = (const float*)d_in[5]; const float* bv = (const float*)d_in[6];
    const float* Wo = (const float*)d_in[7]; const float* bo = (const float*)d_in[8];
    float* out = (float*)d_out;

    const size_t qsz = (size_t)Bb * Hh * Tt * DH;   // elements per tensor
    float* qb = (float*)d_ws;
    float* kb = qb + qsz;
    float* vb = kb + qsz;
    float* ao = vb + qsz;                           // attention output (B,T,C)

    // 1) QKV projections (Q pre-scaled by dh^-0.5)
    dim3 g1(Cc / 64, (ROWS + 15) / 16, 3);
    qkv_proj_kernel<<<g1, 32, 0, stream>>>(X, Wq, bq, Wk, bk, Wv, bv, qb, kb, vb);

    // 2) Local windowed attention
    const size_t smem = (size_t)256 * SSTRIDE * sizeof(float);
    local_attn_kernel<<<dim3(Bb * Hh * Nn), 256, smem, stream>>>(qb, kb, vb, ao);

    // 3) CLS global attention
    cls_attn_kernel<<<dim3(Bb * Hh), 256, 0, stream>>>(qb, kb, vb, ao);

    // 4) Output projection
    out_proj_kernel<<<dim3(Cc / 64, (ROWS + 15) / 16), 32, 0, stream>>>(ao, Wo, bo, out);
}